// GraphFeatureTokenizer_79628693668251
// MI455X (gfx1250) — compile-verified
//
#include <hip/hip_runtime.h>
#include <hip/hip_bf16.h>

typedef __attribute__((ext_vector_type(2))) float v2f;
typedef __attribute__((ext_vector_type(8))) float v8f;

#define B_ 16
#define N_ 2048
#define E_ 4096
#define K_ 32
#define D_ 768
#define T_ (N_ + E_)          // 6144 tokens per graph
#define DCHUNK 128            // columns per wave
#define NSUB (DCHUNK / 16)    // 8 wmma subtiles per wave
#define NDC (D_ / DCHUNK)     // 6 column chunks
#define TT (T_ / 16)          // 384 token tiles per batch

__global__ __launch_bounds__(256)
void GraphFeatureTokenizer_wmma_kernel(const int* __restrict__ edge_data,
                                       const int* __restrict__ node_data,
                                       const int* __restrict__ padded_index,
                                       const float* __restrict__ lap_eigvec,
                                       const float* __restrict__ emb_table,
                                       const float* __restrict__ lap_w,
                                       const float* __restrict__ order_emb,
                                       float* __restrict__ out)
{
    const int wave = (int)((blockIdx.x * blockDim.x + threadIdx.x) >> 5);
    const int lane = (int)(threadIdx.x & 31);
    const int r = lane & 15;   // A row / B col / C col index
    const int h = lane >> 4;   // lane half selects K sub-pair / C row half

    const int per_batch = TT * NDC;
    const int b   = wave / per_batch;
    int       rem = wave - b * per_batch;
    const int tt  = rem / NDC;
    const int dc  = rem - tt * NDC;
    const int t0  = tt * 16;        // first token of this tile
    const int d0  = dc * DCHUNK;    // first output column of this chunk

    // ---- A fragments: x[r][k] = concat(lap[idx0[t0+r]], lap[idx1[t0+r]]) ----
    // 16x4 f32 A layout: lanes 0-15 hold (K=4kk+0, 4kk+1), lanes 16-31 hold
    // (K=4kk+2, 4kk+3) for row M = lane&15  ->  contiguous even float2 loads.
    // lap_eigvec is only 4 MB: keep it regular-temporal so it lives in L2.
    const int2 pidxA = ((const int2*)padded_index)[b * T_ + t0 + r];
    const float* rowA0 = lap_eigvec + (size_t)(b * N_ + pidxA.x) * K_;
    const float* rowA1 = lap_eigvec + (size_t)(b * N_ + pidxA.y) * K_;

    v2f a_frag[16];
#pragma unroll
    for (int kk = 0; kk < 16; ++kk) {
        const int k0 = kk * 4 + 2 * h;  // even, pair never straddles the K_=32 split
        const float* src = (k0 < K_) ? (rowA0 + k0) : (rowA1 + (k0 - K_));
        a_frag[kk] = *(const v2f*)src;
    }

    // ---- order bits for this half's 8 epilogue tokens (m = v + 8h) ----
    int ordmask = 0;
#pragma unroll
    for (int v = 0; v < 8; ++v) {
        const int2 p = ((const int2*)padded_index)[b * T_ + t0 + v + 8 * h];
        ordmask |= (int)(p.x == p.y) << v;
    }

    // ---- token-data base: tile is uniformly node (t0 < N) or edge tokens ----
    const int* dat = (t0 < N_)
        ? (node_data + ((size_t)b * N_ + (size_t)t0) * D_)
        : (edge_data + ((size_t)b * E_ + (size_t)(t0 - N_)) * D_);

    float* outp = out + ((size_t)b * T_ + (size_t)t0) * D_;

#pragma unroll
    for (int s = 0; s < NSUB; ++s) {
        const int dcol = d0 + s * 16 + r;                 // this lane's column
        const float* wrow = lap_w + (size_t)dcol * (2 * K_);

        // B 4x16 layout mirrors A: lanes 0-15 -> (K=4kk+0,+1), 16-31 -> (+2,+3)
        v8f c = {};
#pragma unroll
        for (int kk = 0; kk < 16; ++kk) {
            const int k0 = kk * 4 + 2 * h;
            const v2f bfrag = *(const v2f*)(wrow + k0);
            c = __builtin_amdgcn_wmma_f32_16x16x4_f32(
                    /*neg_a=*/false, a_frag[kk],
                    /*neg_b=*/false, bfrag,
                    /*c_mod=*/(short)0, c,
                    /*reuse_a=*/false, /*reuse_b=*/false);
        }

        // C layout: c[v] is token m = v + 8h, column n = lane&15.
        // Fused epilogue: embedding gather + order embedding + store.
        // Token data is read once and output written once (~604 MB streamed):
        // use non-temporal policy so the hot lap/emb tables stay resident in L2.
#pragma unroll
        for (int v = 0; v < 8; ++v) {
            const int m   = v + 8 * h;
            const int tok = __builtin_nontemporal_load(dat + (size_t)m * D_ + dcol);
            const float feat = emb_table[tok];            // 4 KB table, L0/L2 hot
            const int  ord = (ordmask >> v) & 1;
            const float oe = order_emb[ord * D_ + dcol];
            __builtin_nontemporal_store(c[v] + feat + oe,
                                        outp + (size_t)m * D_ + dcol);
        }
    }
}

extern "C" void kernel_launch(void* const* d_in, const int* in_sizes, int n_in,
                              void* d_out, int out_size, void* d_ws, size_t ws_size,
                              hipStream_t stream) {
    (void)in_sizes; (void)n_in; (void)d_ws; (void)ws_size; (void)out_size;
    // setup_inputs order:
    // 0 edge_index, 1 edge_data, 2 node_data, 3 node_num, 4 edge_num,
    // 5 padded_index, 6 padding_mask, 7 padded_node_mask, 8 padded_edge_mask,
    // 9 lap_eigvec, 10 emb_table, 11 lap_w, 12 order_emb
    const int*   edge_data    = (const int*)  d_in[1];
    const int*   node_data    = (const int*)  d_in[2];
    const int*   padded_index = (const int*)  d_in[5];
    const float* lap_eigvec   = (const float*)d_in[9];
    const float* emb_table    = (const float*)d_in[10];
    const float* lap_w        = (const float*)d_in[11];
    const float* order_emb    = (const float*)d_in[12];
    float*       out          = (float*)d_out;

    const int total_waves = B_ * TT * NDC;        // 16 * 384 * 6 = 36864
    const int threads     = 256;                  // 8 waves per block
    const int blocks      = (total_waves * 32) / threads;  // 4608

    GraphFeatureTokenizer_wmma_kernel<<<blocks, threads, 0, stream>>>(
        edge_data, node_data, padded_index, lap_eigvec,
        emb_table, lap_w, order_emb, out);
}